// GConv_67688684585519
// MI455X (gfx1250) — compile-verified
//
#include <hip/hip_runtime.h>
#include <hip/hip_bf16.h>

typedef __attribute__((ext_vector_type(2))) float v2f;
typedef __attribute__((ext_vector_type(8))) float v8f;

// ---------------- utility: zero a float buffer ----------------
__global__ void gcn_zero_f32(float* __restrict__ p, int n) {
    int i = blockIdx.x * blockDim.x + threadIdx.x;
    if (i < n) p[i] = 0.0f;
}

// ---------------- degree: deg[col[e]] += 1 ----------------
__global__ void gcn_degree(const int* __restrict__ col, float* __restrict__ deg, int nedges) {
    int e = blockIdx.x * blockDim.x + threadIdx.x;
    if (e < nedges) atomicAdd(&deg[col[e]], 1.0f);
}

// ---------------- dinv = rsqrt(deg + 1)  (self loop contributes 1; always > 0) ----------------
__global__ void gcn_dinv(float* __restrict__ deg, int n) {
    int i = blockIdx.x * blockDim.x + threadIdx.x;
    if (i < n) deg[i] = rsqrtf(deg[i] + 1.0f);
}

// ---------------- GEMM: Z[nrows x 64] = X[nrows x 64] @ W[64 x 64] ----------------
// One wave per 16-row strip; computes the full 16x64 output (4 col-tiles) so each
// A fragment feeds 4 WMMAs and x is read exactly once. 64 V_WMMA_F32_16X16X4_F32
// per wave, K accumulated in 16 steps of 4.
// A 16x4 f32 layout (ISA 7.12.2): lanes 0-15 hold M=lane, K={0,1} in v0/v1;
// lanes 16-31 hold M=lane-16, K={2,3}. B 4x16 symmetric (N in place of M).
// C/D 16x16 f32: VGPR r -> M = r + (lane>=16 ? 8 : 0), N = lane&15.
__global__ void gcn_gemm64_wmma(const float* __restrict__ X, const float* __restrict__ W,
                                float* __restrict__ Z, int nrows) {
    const int wavesPerBlock = blockDim.x >> 5;
    const int waveId = blockIdx.x * wavesPerBlock + (threadIdx.x >> 5);
    const int lane = threadIdx.x & 31;
    const int nRowTiles = (nrows + 15) >> 4;
    if (waveId >= nRowTiles) return;                // wave-uniform: EXEC stays all-ones
    const int m0 = waveId << 4;
    const int mr = lane & 15;                       // M for A / N for B,C
    const int kk = (lane >> 4) << 1;                // 0 or 2: K sub-offset per half-wave

    int mLoad = m0 + mr;
    if (mLoad > nrows - 1) mLoad = nrows - 1;       // clamp (stores are guarded)
    const float* xrow = X + (size_t)mLoad * 64;

    v8f c0 = {}, c1 = {}, c2 = {}, c3 = {};
#pragma unroll
    for (int k0 = 0; k0 < 64; k0 += 4) {
        v2f a, b;
        a.x = xrow[k0 + kk];
        a.y = xrow[k0 + kk + 1];
        const float* w0 = W + (k0 + kk) * 64 + mr;      // K = k0+kk   row
        const float* w1 = w0 + 64;                      // K = k0+kk+1 row
        // 8 args: (neg_a, A, neg_b, B, c_mod, C, reuse_a, reuse_b)
        b.x = w0[0];  b.y = w1[0];
        c0 = __builtin_amdgcn_wmma_f32_16x16x4_f32(false, a, false, b, (short)0, c0, false, false);
        b.x = w0[16]; b.y = w1[16];
        c1 = __builtin_amdgcn_wmma_f32_16x16x4_f32(false, a, false, b, (short)0, c1, false, false);
        b.x = w0[32]; b.y = w1[32];
        c2 = __builtin_amdgcn_wmma_f32_16x16x4_f32(false, a, false, b, (short)0, c2, false, false);
        b.x = w0[48]; b.y = w1[48];
        c3 = __builtin_amdgcn_wmma_f32_16x16x4_f32(false, a, false, b, (short)0, c3, false, false);
    }

    const int mbase = m0 + ((lane >> 4) << 3);      // +8 for upper half-wave
#pragma unroll
    for (int r = 0; r < 8; ++r) {
        int m = mbase + r;
        if (m < nrows) {
            float* zr = Z + (size_t)m * 64 + mr;
            zr[0]  = c0[r];
            zr[16] = c1[r];
            zr[32] = c2[r];
            zr[48] = c3[r];
        }
    }
}

// ---------------- edge scatter: H[col] += Z[row] * dinv[row]*dinv[col] ----------------
// Two edges per wave: each 16-lane half handles one edge with a float4 per lane
// -> one coalesced 256B global_load_b128 burst per edge, 64 global_atomic_add_f32
// per edge. Working set (z,h ~51MB) is resident in MI455X's 192MB L2, so the
// atomics resolve in L2 rather than HBM.
__global__ void gcn_scatter(const float* __restrict__ Z, const int* __restrict__ row,
                            const int* __restrict__ col, const float* __restrict__ dinv,
                            float* __restrict__ H, int nedges) {
    const int lane = threadIdx.x & 31;
    const int half = lane >> 4;                     // which edge within the wave
    const int l = lane & 15;                        // feature-quad index
    const int e = (blockIdx.x * (blockDim.x >> 5) + (threadIdx.x >> 5)) * 2 + half;
    if (e >= nedges) return;
    const int r = row[e];
    const int c = col[e];
    const float w = dinv[r] * dinv[c];
    const float4 v = ((const float4*)(Z + (size_t)r * 64))[l];
    float* hp = H + (size_t)c * 64 + 4 * l;
    atomicAdd(hp + 0, v.x * w);
    atomicAdd(hp + 1, v.y * w);
    atomicAdd(hp + 2, v.z * w);
    atomicAdd(hp + 3, v.w * w);
}

// ---------------- fused self-loop + bias + relu ----------------
// Hout[i,d] = relu(Hin[i,d] + Z[i,d]*dinv[i]^2 + bias[d]); the kernel boundary after
// the scatter makes the non-atomic self-loop add safe. Hin may alias Hout.
__global__ void gcn_fuse(const float* __restrict__ Z, const float* __restrict__ dinv,
                         const float* __restrict__ bias, const float* __restrict__ Hin,
                         float* __restrict__ Hout, int nnodes) {
    const int idx = blockIdx.x * blockDim.x + threadIdx.x;   // node*16 + quad
    const int node = idx >> 4;
    if (node >= nnodes) return;
    const int d = (idx & 15) * 4;
    const float di = dinv[node];
    const float w = di * di;
    const size_t off = (size_t)node * 64 + d;
    const float4 z = *(const float4*)(Z + off);
    const float4 h = *(const float4*)(Hin + off);
    const float4 b = *(const float4*)(bias + d);
    float4 o;
    o.x = fmaxf(fmaf(z.x, w, h.x) + b.x, 0.0f);
    o.y = fmaxf(fmaf(z.y, w, h.y) + b.y, 0.0f);
    o.z = fmaxf(fmaf(z.z, w, h.z) + b.z, 0.0f);
    o.w = fmaxf(fmaf(z.w, w, h.w) + b.w, 0.0f);
    *(float4*)(Hout + off) = o;
}

extern "C" void kernel_launch(void* const* d_in, const int* in_sizes, int n_in,
                              void* d_out, int out_size, void* d_ws, size_t ws_size,
                              hipStream_t stream) {
    const float* x  = (const float*)d_in[0];
    const int*   ei = (const int*)d_in[1];       // (2, E) row-major
    const float* W1 = (const float*)d_in[2];
    const float* b1 = (const float*)d_in[3];
    const float* W2 = (const float*)d_in[4];
    const float* b2 = (const float*)d_in[5];

    const int N = in_sizes[0] / 64;
    const int E = in_sizes[1] / 2;
    const int* row = ei;
    const int* col = ei + E;

    // workspace: dinv[N] | z[N*64] | h[N*64]  (~51.6 MB for N=100k)
    float* dinv = (float*)d_ws;
    float* zbuf = dinv + (size_t)N;
    float* hbuf = zbuf + (size_t)N * 64;
    float* out  = (float*)d_out;

    const int NT = 256;
    const int nFeat = N * 64;
    const int gemmTiles = (N + 15) / 16;              // one wave per 16-row strip
    const int gemmBlocks = (gemmTiles + 7) / 8;       // 8 waves / 256-thread block
    const int scatBlocks = (E + 15) / 16;             // 16 edges / block (2 per wave)
    const int fuseBlocks = (N * 16 + NT - 1) / NT;

    // normalization
    gcn_zero_f32<<<(N + NT - 1) / NT, NT, 0, stream>>>(dinv, N);
    gcn_degree<<<(E + NT - 1) / NT, NT, 0, stream>>>(col, dinv, E);
    gcn_dinv<<<(N + NT - 1) / NT, NT, 0, stream>>>(dinv, N);

    // layer 1
    gcn_gemm64_wmma<<<gemmBlocks, NT, 0, stream>>>(x, W1, zbuf, N);
    gcn_zero_f32<<<(nFeat + NT - 1) / NT, NT, 0, stream>>>(hbuf, nFeat);
    gcn_scatter<<<scatBlocks, NT, 0, stream>>>(zbuf, row, col, dinv, hbuf, E);
    gcn_fuse<<<fuseBlocks, NT, 0, stream>>>(zbuf, dinv, b1, hbuf, hbuf, N);

    // layer 2
    gcn_gemm64_wmma<<<gemmBlocks, NT, 0, stream>>>(hbuf, W2, zbuf, N);
    gcn_zero_f32<<<(nFeat + NT - 1) / NT, NT, 0, stream>>>(out, nFeat);
    gcn_scatter<<<scatBlocks, NT, 0, stream>>>(zbuf, row, col, dinv, out, E);
    gcn_fuse<<<fuseBlocks, NT, 0, stream>>>(zbuf, dinv, b2, out, out, N);
}